// RoIPooling3D_80109730005436
// MI455X (gfx1250) — compile-verified
//
#include <hip/hip_runtime.h>

// ---------------------------------------------------------------------------
// 3D adaptive RoI max-pool for MI455X (gfx1250).
// Strategy: one workgroup per (roi, channel). The RoI sub-volume
// (<=16x32x32 f32, rows <=128B with stride 256B) is staged into LDS with ONE
// Tensor-Data-Mover instruction (tensor_load_to_lds, 3D descriptor), then 98
// lanes max-reduce their adaptive-pool bins out of the packed LDS tile.
// Features (128 MiB) are L2-resident (192MB L2), so tile traffic is L2->LDS.
// ---------------------------------------------------------------------------

typedef unsigned int       u32;
typedef unsigned long long u64;
typedef __attribute__((ext_vector_type(4))) u32 u32x4;
typedef __attribute__((ext_vector_type(8))) u32 u32x8;

#define C_  256
#define T_  16
#define H_  64
#define W_  64
#define N_  128
#define OT_ 2
#define OH_ 7
#define OW_ 7
#define OBINS_ (OT_ * OH_ * OW_)   // 98

__global__ __launch_bounds__(128) void roipool3d_tdm_kernel(
    const float* __restrict__ feat,   // [B=2, C, T, H, W]
    const float* __restrict__ rois,   // [N, 7] = b, t1, x1, y1, t2, x2, y2
    float* __restrict__ out)          // [N, C, OT, OH, OW]
{
  // Worst-case tile: 16 x 32 x 32 f32 = 64 KB (packed by the TDM).
  __shared__ float tile[T_ * 32 * 32];

  const u32 nc = blockIdx.x;
  const u32 n  = nc >> 8;      // / C_
  const u32 c  = nc & 255u;    // % C_

  // RoI coords: block-uniform scalar loads. TS=SS=1.0 and values are
  // non-negative integers stored as floats, so (int) == floor(v * scale).
  const float* __restrict__ r = rois + (u64)n * 7;
  const int b  = (int)r[0];
  const int t1 = (int)r[1];
  const int x1 = (int)r[2];
  const int y1 = (int)r[3];
  const int tl = (int)r[4] - t1 + 1;   // 4..16
  const int wl = (int)r[5] - x1 + 1;   // 8..32
  const int hl = (int)r[6] - y1 + 1;   // 8..32

  // ---- Phase 1: wave 0 issues one TDM 3D tile load global -> LDS ----------
  if (threadIdx.x < 32) {
    const u64 ga  = (u64)(feat +
        ((((u64)b * C_ + c) * T_ + (u32)t1) * H_ + (u32)y1) * W_ + (u32)x1);
    const u32 lds = (u32)(u64)(&tile[0]);   // low 32 bits = LDS byte offset

    // D# group 0 (128b): count=1 | lds_addr | global_addr[56:0] | type=2
    u32x4 g0;
    g0[0] = 1u;                                              // count=1, user
    g0[1] = lds;                                             // lds_addr
    g0[2] = (u32)ga;                                         // gaddr[31:0]
    g0[3] = ((u32)(ga >> 32) & 0x01FFFFFFu) | (2u << 30);    // gaddr[56:32]|type=2

    // D# group 1 (256b): data_size=4B; tensor dims = exact tile extents
    // (tile fully in-bounds => no OOB zero-fill under any interpretation);
    // strides: dim0 = W elements, dim1 = H*W elements.
    const u32 td0 = (u32)wl, td1 = (u32)hl;
    const u32 s0  = W_;            // 64
    const u32 s1  = H_ * W_;       // 4096
    u32x8 g1;
    g1[0] = (2u << 16);                                // mask=0, data_size=4B
    g1[1] = (td0 & 0xFFFFu) << 16;                     // barrier_addr=0 | dim0.lo
    g1[2] = (td0 >> 16) | ((td1 & 0xFFFFu) << 16);     // dim0.hi | dim1.lo
    g1[3] = (td1 >> 16) | ((u32)wl << 16);             // dim1.hi | tile_dim0
    g1[4] = (u32)hl | ((u32)tl << 16);                 // tile_dim1 | tile_dim2
    g1[5] = s0;                                        // dim0_stride[31:0]
    g1[6] = (s1 & 0xFFFFu) << 16;                      // stride0.hi=0|stride1.lo
    g1[7] = s1 >> 16;                                  // stride1[47:16]

    // D# group 2: tensor_dim2 = tl; 4th dim unused (tile_dim3 = 0).
    u32x4 g2;
    g2[0] = (u32)tl;
    g2[1] = 0u;
    g2[2] = 0u;
    g2[3] = 0u;

    // D# group 3: 5th dim unused.
    u32x4 g3;
    g3[0] = 0u; g3[1] = 0u; g3[2] = 0u; g3[3] = 0u;

    asm volatile("tensor_load_to_lds %0, %1, %2, %3"
                 :
                 : "s"(g0), "s"(g1), "s"(g2), "s"(g3)
                 : "memory");
#if __has_builtin(__builtin_amdgcn_s_wait_tensorcnt)
    __builtin_amdgcn_s_wait_tensorcnt(0);
#else
    asm volatile("s_wait_tensorcnt 0x0" ::: "memory");
#endif
  }
  __syncthreads();   // make TDM-written LDS visible to all 4 waves

  // ---- Phase 2: 98 lanes, one adaptive-pool bin each ----------------------
  const u32 tid = threadIdx.x;
  if (tid < OBINS_) {
    const int ow = (int)(tid % OW_);
    const int oh = (int)((tid / OW_) % OH_);
    const int ot = (int)(tid / (OW_ * OH_));

    // bin = [s, e) with s = o*L/O, e = ceil((o+1)*L/O)  (matches reference)
    const int ts = (ot * tl) / OT_;
    const int te = ((ot + 1) * tl + OT_ - 1) / OT_;
    const int ys = (oh * hl) / OH_;
    const int ye = ((oh + 1) * hl + OH_ - 1) / OH_;
    const int xs = (ow * wl) / OW_;
    const int xe = ((ow + 1) * wl + OW_ - 1) / OW_;

    float m = -__builtin_inff();
    for (int t = ts; t < te; ++t) {
      const float* __restrict__ pt = &tile[(u32)(t * hl) * (u32)wl];
      for (int y = ys; y < ye; ++y) {
        const float* __restrict__ py = pt + (u32)y * (u32)wl;
        for (int x = xs; x < xe; ++x) {
          m = fmaxf(m, py[x]);
        }
      }
    }
    // flat bin index == tid, so output offset is (n*C + c)*98 + tid
    out[(u64)nc * OBINS_ + tid] = m;
  }
}

extern "C" void kernel_launch(void* const* d_in, const int* in_sizes, int n_in,
                              void* d_out, int out_size, void* d_ws, size_t ws_size,
                              hipStream_t stream) {
  (void)in_sizes; (void)n_in; (void)out_size; (void)d_ws; (void)ws_size;
  const float* feat = (const float*)d_in[0];   // [2,256,16,64,64] f32
  const float* rois = (const float*)d_in[1];   // [128,7] f32
  float* out = (float*)d_out;                  // [128,256,2,7,7] f32

  dim3 grid(N_ * C_);   // 32768 blocks: one per (roi, channel)
  dim3 block(128);      // 4 wave32s
  roipool3d_tdm_kernel<<<grid, block, 0, stream>>>(feat, rois, out);
}